// CPDP_POMO_net_7009386627333
// MI455X (gfx1250) — compile-verified
//
#include <hip/hip_runtime.h>
#include <math.h>

// Problem constants (from reference): B=1024, N=500, D=128, H=8, L=2, CLIP=10
#define BSZ 1024
#define NK  500
#define DD  128
#define HH  8
#define ROWSTRIDE 256          // K_att / V_att innermost stride in floats (D*L)

typedef __attribute__((ext_vector_type(2))) float v2f;
typedef __attribute__((ext_vector_type(8))) float v8f;

__device__ __forceinline__ float waveReduceSum(float v) {
#pragma unroll
  for (int o = 16; o > 0; o >>= 1) v += __shfl_xor(v, o, 32);
  return v;
}

// dst[r] = sum_k W[r,k] * vec[k] + bias[r], r = 0..127, via V_WMMA_F32_16X16X4_F32.
// Wave w computes rows 16w..16w+15. B operand = vec broadcast across all 16 columns
// (redundant columns are free; kernel is HBM-bound). A layout (16x4 f32): lanes 0-15
// hold M rows with K={0,1} in the two VGPRs, lanes 16-31 hold K={2,3}. D layout:
// VGPR j -> row j (lanes 0-15) / row 8+j (lanes 16-31).
__device__ __forceinline__ void matvec_wmma(const float* __restrict__ W,
                                            const float* __restrict__ bias,
                                            const float* __restrict__ vec_s,
                                            float* __restrict__ dst_s,
                                            int wave, int lane) {
  const int r0 = wave * 16;
  const int m  = lane & 15;
  const int kh = (lane >> 4) * 2;             // 0 or 2 (K offset for this half-wave)
  const float* wrow = W + (size_t)(r0 + m) * DD;
  v8f acc = {};
#pragma unroll 4
  for (int kb = 0; kb < DD; kb += 4) {
    v2f a, bv;
    a.x  = wrow[kb + kh + 0];
    a.y  = wrow[kb + kh + 1];
    bv.x = vec_s[kb + kh + 0];                // B[k,n] = vec[k] for every column n
    bv.y = vec_s[kb + kh + 1];
    acc = __builtin_amdgcn_wmma_f32_16x16x4_f32(false, a, false, bv,
                                                (short)0, acc, false, false);
  }
  if (m == 0) {                               // lane 0 -> rows r0..r0+7, lane 16 -> r0+8..r0+15
    const int rbase = r0 + (lane >> 4) * 8;
#pragma unroll
    for (int j = 0; j < 8; ++j) dst_s[rbase + j] = acc[j] + bias[rbase + j];
  }
}

__global__ __launch_bounds__(256)
void fused_two_layer_attn(const float* __restrict__ query,   // (B,1,128)
                          const float* __restrict__ K_att,   // (B,500,256)
                          const float* __restrict__ V_att,   // (B,500,256)
                          const unsigned char* __restrict__ mask, // (B,500) bool
                          const float* __restrict__ W0_w, const float* __restrict__ W0_b,
                          const float* __restrict__ Wq_w, const float* __restrict__ Wq_b,
                          float* __restrict__ out) {         // (B,500)
  __shared__ float sc_s[HH * NK];   // layer-0 scores/weights (8x500); first 500 reused in phase 3
  __shared__ float q_s[DD];
  __shared__ float o_s[DD];         // layer-0 attention output
  __shared__ float q1_s[DD];
  __shared__ float qf_s[DD];
  __shared__ float red_s[8];

  const int b    = blockIdx.x;
  const int tid  = threadIdx.x;
  const int wave = tid >> 5;
  const int lane = tid & 31;

  if (tid < DD) { q_s[tid] = query[(size_t)b * DD + tid]; o_s[tid] = 0.0f; }
  __syncthreads();

  const float* Kb = K_att + (size_t)b * NK * ROWSTRIDE;
  const float* Vb = V_att + (size_t)b * NK * ROWSTRIDE;
  const unsigned char* mb = mask + (size_t)b * NK;

  // ---------- Phase 1a: layer-0 scores. Wave per key row; lane l covers d=4l..4l+3;
  // lane group of 4 == one head (dh=16). Coalesced 512B row loads. ----------
  {
    const float q0 = q_s[4*lane+0], q1v = q_s[4*lane+1],
                q2 = q_s[4*lane+2], q3  = q_s[4*lane+3];
    for (int n = wave; n < NK; n += 8) {
      const float4 k4 = *(const float4*)(Kb + (size_t)n * ROWSTRIDE + 4*lane);
      if (n + 16 < NK)
        __builtin_prefetch(Kb + (size_t)(n + 16) * ROWSTRIDE + 4*lane, 0, 1);
      float p = q0*k4.x + q1v*k4.y + q2*k4.z + q3*k4.w;
      p += __shfl_xor(p, 1, 32);
      p += __shfl_xor(p, 2, 32);                      // reduce 4-lane head group
      float s = p * 0.25f;                            // 1/sqrt(16)
      if (mb[n]) s = -1.0e9f;
      if ((lane & 3) == 0) sc_s[(lane >> 2) * NK + n] = s;
    }
  }
  __syncthreads();

  // ---------- Phase 1b: per-head masked softmax; wave h owns head h ----------
  {
    float* sh = sc_s + wave * NK;
    float m = -INFINITY;
    for (int n = lane; n < NK; n += 32) m = fmaxf(m, sh[n]);
#pragma unroll
    for (int o = 16; o > 0; o >>= 1) m = fmaxf(m, __shfl_xor(m, o, 32));
    float sum = 0.0f;
    for (int n = lane; n < NK; n += 32) {
      float e = __expf(sh[n] - m); sh[n] = e; sum += e;
    }
    sum = waveReduceSum(sum);
    const float inv = 1.0f / sum;
    for (int n = lane; n < NK; n += 32) sh[n] *= inv;
  }
  __syncthreads();

  // ---------- Phase 1c: out = w @ V, accumulated per wave, merged via ds_add_f32 ----------
  {
    const float* wh = sc_s + (lane >> 2) * NK;        // this lane's head weights
    float a0 = 0, a1 = 0, a2 = 0, a3 = 0;
    for (int n = wave; n < NK; n += 8) {
      const float4 v4 = *(const float4*)(Vb + (size_t)n * ROWSTRIDE + 4*lane);
      const float wgt = wh[n];
      a0 += wgt * v4.x; a1 += wgt * v4.y; a2 += wgt * v4.z; a3 += wgt * v4.w;
    }
    atomicAdd(&o_s[4*lane+0], a0);
    atomicAdd(&o_s[4*lane+1], a1);
    atomicAdd(&o_s[4*lane+2], a2);
    atomicAdd(&o_s[4*lane+3], a3);
  }
  __syncthreads();

  // ---------- Phase 2: qf = Wq @ (W0 @ out + b0) + bq using f32 WMMA ----------
  matvec_wmma(W0_w, W0_b, o_s, q1_s, wave, lane);
  __syncthreads();
  matvec_wmma(Wq_w, Wq_b, q1_s, qf_s, wave, lane);
  __syncthreads();

  // ---------- Phase 3: layer-1 scores (single head, d=128), clip = 10*tanh ----------
  {
    const float f0 = qf_s[4*lane+0], f1 = qf_s[4*lane+1],
                f2 = qf_s[4*lane+2], f3 = qf_s[4*lane+3];
    for (int n = wave; n < NK; n += 8) {
      const float4 k4 = *(const float4*)(Kb + (size_t)n * ROWSTRIDE + DD + 4*lane);
      if (n + 16 < NK)
        __builtin_prefetch(Kb + (size_t)(n + 16) * ROWSTRIDE + DD + 4*lane, 0, 1);
      float p = f0*k4.x + f1*k4.y + f2*k4.z + f3*k4.w;
      p = waveReduceSum(p);
      float s = 10.0f * tanhf(p * 0.08838834764831845f);  // 1/sqrt(128)
      if (mb[n]) s = -1.0e9f;
      if (lane == 0) sc_s[n] = s;
    }
  }
  __syncthreads();

  // ---------- Phase 3b: block-wide softmax over 500, write attention weights ----------
  {
    float m = -INFINITY;
    for (int n = tid; n < NK; n += 256) m = fmaxf(m, sc_s[n]);
#pragma unroll
    for (int o = 16; o > 0; o >>= 1) m = fmaxf(m, __shfl_xor(m, o, 32));
    if (lane == 0) red_s[wave] = m;
    __syncthreads();
    m = fmaxf(fmaxf(fmaxf(red_s[0], red_s[1]), fmaxf(red_s[2], red_s[3])),
              fmaxf(fmaxf(red_s[4], red_s[5]), fmaxf(red_s[6], red_s[7])));
    __syncthreads();                                   // red_s reuse hazard guard

    float sum = 0.0f;
    for (int n = tid; n < NK; n += 256) {
      float e = __expf(sc_s[n] - m); sc_s[n] = e; sum += e;
    }
    sum = waveReduceSum(sum);
    if (lane == 0) red_s[wave] = sum;
    __syncthreads();
    sum = red_s[0] + red_s[1] + red_s[2] + red_s[3]
        + red_s[4] + red_s[5] + red_s[6] + red_s[7];
    const float inv = 1.0f / sum;
    float* outb = out + (size_t)b * NK;
    for (int n = tid; n < NK; n += 256) outb[n] = sc_s[n] * inv;
  }
}

extern "C" void kernel_launch(void* const* d_in, const int* in_sizes, int n_in,
                              void* d_out, int out_size, void* d_ws, size_t ws_size,
                              hipStream_t stream) {
  const float*         query = (const float*)d_in[0];
  const float*         K_att = (const float*)d_in[1];
  const float*         V_att = (const float*)d_in[2];
  const unsigned char* mask  = (const unsigned char*)d_in[3]; // jax bool = 1 byte/elem
  const float*         W0_w  = (const float*)d_in[4];
  const float*         W0_b  = (const float*)d_in[5];
  const float*         Wq_w  = (const float*)d_in[6];
  const float*         Wq_b  = (const float*)d_in[7];

  fused_two_layer_attn<<<BSZ, 256, 0, stream>>>(
      query, K_att, V_att, mask, W0_w, W0_b, Wq_w, Wq_b, (float*)d_out);
}